// N3TreeEstimator_1726576856127
// MI455X (gfx1250) — compile-verified
//
#include <hip/hip_runtime.h>
#include <stdint.h>

#define BLOCK 256
#define ITEMS 8
#define CHUNK (BLOCK * ITEMS)
#define EPS 1.0e-4f

// LDS byte-offset of a shared-memory pointer (addrspacecast generic->local,
// ptrtoint of the 32-bit local pointer == offset within the group segment).
__device__ __forceinline__ unsigned lds_off_u32(const void* p) {
  return (unsigned)(unsigned long long)(const __attribute__((address_space(3))) char*)p;
}

// Segmented-product monoid: (p,f) (+) (q,g) = (g ? q : p*q, f|g).
// Hillis-Steele inclusive scan over BLOCK entries; returns exclusive prefix.
__device__ __forceinline__ void block_scan(float& p, unsigned& f,
                                           float* sp, unsigned* sf,
                                           float& ep, unsigned& ef) {
  const int t = threadIdx.x;
  sp[t] = p; sf[t] = f;
  __syncthreads();
#pragma unroll
  for (int d = 1; d < BLOCK; d <<= 1) {
    float lp = 1.0f; unsigned lf = 0u;
    const bool act = (t >= d);
    if (act) { lp = sp[t - d]; lf = sf[t - d]; }
    __syncthreads();
    if (act) {
      p = f ? p : lp * p;
      f |= lf;
      sp[t] = p; sf[t] = f;
    }
    __syncthreads();
  }
  if (t == 0) { ep = 1.0f; ef = 0u; }
  else        { ep = sp[t - 1]; ef = sf[t - 1]; }
  __syncthreads();
}

// ---------------- Pass 1: per-chunk aggregates ----------------
__global__ __launch_bounds__(BLOCK)
void k_aggregate(const float* __restrict__ alphas,
                 const int* __restrict__ ridx,
                 int n,
                 float* __restrict__ aggP,
                 unsigned* __restrict__ aggF) {
  __shared__ float sp[BLOCK];
  __shared__ unsigned sf[BLOCK];
  const int t    = threadIdx.x;
  const int i0   = (int)blockIdx.x * CHUNK;
  const int base = i0 + t * ITEMS;

  // gfx1250 prefetch of the next chunk's stream (global_prefetch_b8)
  if (i0 + CHUNK + t * ITEMS < n) {
    __builtin_prefetch(alphas + i0 + CHUNK + t * ITEMS, 0, 1);
    __builtin_prefetch(ridx   + i0 + CHUNK + t * ITEMS, 0, 1);
  }

  float p = 1.0f; unsigned f = 0u;
  if (base < n) {
    float a[ITEMS]; int id[ITEMS];
    const bool tf = (base + ITEMS) <= n;
    if (tf) {
      const float4* ap = (const float4*)(alphas + base);
      float4 a0 = ap[0], a1 = ap[1];
      a[0]=a0.x; a[1]=a0.y; a[2]=a0.z; a[3]=a0.w;
      a[4]=a1.x; a[5]=a1.y; a[6]=a1.z; a[7]=a1.w;
      const int4* rp = (const int4*)(ridx + base);
      int4 r0 = rp[0], r1 = rp[1];
      id[0]=r0.x; id[1]=r0.y; id[2]=r0.z; id[3]=r0.w;
      id[4]=r1.x; id[5]=r1.y; id[6]=r1.z; id[7]=r1.w;
    } else {
#pragma unroll
      for (int k = 0; k < ITEMS; ++k) {
        int i = base + k;
        a[k]  = (i < n) ? alphas[i] : 0.0f;
        id[k] = (i < n) ? ridx[i]   : 0;
      }
    }
    int prev = (base > 0) ? ridx[base - 1] : (int)0x80000000;
#pragma unroll
    for (int k = 0; k < ITEMS; ++k) {
      int i = base + k;
      if (i < n) {
        bool s  = (i == 0) || (id[k] != prev);
        float v = 1.0f - a[k];
        p = s ? v : p * v;
        f |= (unsigned)s;
        prev = id[k];
      }
    }
  }
  float ep; unsigned ef;
  block_scan(p, f, sp, sf, ep, ef);
  (void)ep; (void)ef;
  if (t == BLOCK - 1) { aggP[blockIdx.x] = p; aggF[blockIdx.x] = f; }
}

// ---------------- Pass 2: exclusive scan of block aggregates ----------------
__global__ __launch_bounds__(BLOCK)
void k_carry(const float* __restrict__ aggP,
             const unsigned* __restrict__ aggF,
             int nblocks,
             float* __restrict__ carry) {
  __shared__ float sp[BLOCK];
  __shared__ unsigned sf[BLOCK];
  const int t  = threadIdx.x;
  const int K  = (nblocks + BLOCK - 1) / BLOCK;
  const int b0 = t * K;

  float p = 1.0f; unsigned f = 0u;
  for (int j = 0; j < K; ++j) {
    int b = b0 + j;
    if (b < nblocks) {
      float q = aggP[b]; unsigned g = aggF[b];
      p = g ? q : p * q;
      f |= g;
    }
  }
  float ep; unsigned ef;
  block_scan(p, f, sp, sf, ep, ef);
  float rp = ep; (void)ef;
  for (int j = 0; j < K; ++j) {
    int b = b0 + j;
    if (b < nblocks) {
      carry[b] = rp;                       // incoming transmittance prefix
      float q = aggP[b]; unsigned g = aggF[b];
      rp = g ? q : rp * q;
    }
  }
}

// ---------------- Pass 3: emit trans + vis ----------------
__global__ __launch_bounds__(BLOCK)
void k_final(const float* __restrict__ alphas,
             const int* __restrict__ ridx,
             int n,
             const float* __restrict__ carry,
             float* __restrict__ trans,
             float* __restrict__ vis) {
  __shared__ float s_a[CHUNK];
  __shared__ float sp[BLOCK];
  __shared__ unsigned sf[BLOCK];
  const int t    = threadIdx.x;
  const int i0   = (int)blockIdx.x * CHUNK;
  const int base = i0 + t * ITEMS;
  const bool blk_full = (i0 + CHUNK) <= n;

  if (i0 + CHUNK + t * ITEMS < n)
    __builtin_prefetch(ridx + i0 + CHUNK + t * ITEMS, 0, 1);

  // Stage this lane's 8 alphas LDS-side via CDNA5 async-to-LDS DMA path.
  if (blk_full) {
    unsigned lo = lds_off_u32(&s_a[t * ITEMS]);
    unsigned long long ga = (unsigned long long)(const void*)(alphas + base);
    asm volatile(
        "global_load_async_to_lds_b128 %0, %1, off\n\t"
        "global_load_async_to_lds_b128 %0, %1, off offset:16"
        :: "v"(lo), "v"(ga) : "memory");
  }

  float v[ITEMS]; bool st[ITEMS];
  float p = 1.0f; unsigned f = 0u;

  if (base < n) {
    int id[ITEMS];
    if (blk_full) {
      const int4* rp = (const int4*)(ridx + base);
      int4 r0 = rp[0], r1 = rp[1];
      id[0]=r0.x; id[1]=r0.y; id[2]=r0.z; id[3]=r0.w;
      id[4]=r1.x; id[5]=r1.y; id[6]=r1.z; id[7]=r1.w;
    } else {
#pragma unroll
      for (int k = 0; k < ITEMS; ++k) {
        int i = base + k;
        id[k] = (i < n) ? ridx[i] : 0;
      }
    }
    int prev = (base > 0) ? ridx[base - 1] : (int)0x80000000;

    if (blk_full) {
      asm volatile("s_wait_asynccnt 0x0" ::: "memory");
#pragma unroll
      for (int k = 0; k < ITEMS; ++k) v[k] = 1.0f - s_a[t * ITEMS + k];
    } else {
#pragma unroll
      for (int k = 0; k < ITEMS; ++k) {
        int i = base + k;
        v[k] = (i < n) ? (1.0f - alphas[i]) : 1.0f;
      }
    }
#pragma unroll
    for (int k = 0; k < ITEMS; ++k) {
      int i = base + k;
      if (i < n) {
        bool s = (i == 0) || (id[k] != prev);
        st[k] = s;
        p = s ? v[k] : p * v[k];
        f |= (unsigned)s;
        prev = id[k];
      } else {
        st[k] = false;
      }
    }
  } else {
#pragma unroll
    for (int k = 0; k < ITEMS; ++k) { v[k] = 1.0f; st[k] = false; }
  }

  float ep; unsigned ef;
  block_scan(p, f, sp, sf, ep, ef);

  float run = ef ? ep : carry[blockIdx.x] * ep;

  float tr[ITEMS], vz[ITEMS];
#pragma unroll
  for (int k = 0; k < ITEMS; ++k) {
    if (st[k]) run = 1.0f;
    tr[k] = run;
    vz[k] = (run >= EPS) ? 1.0f : 0.0f;
    run *= v[k];
  }

  if (base + ITEMS <= n) {
    float4* tp = (float4*)(trans + base);
    tp[0] = make_float4(tr[0], tr[1], tr[2], tr[3]);
    tp[1] = make_float4(tr[4], tr[5], tr[6], tr[7]);
    if ((n & 3) == 0) {
      float4* vp = (float4*)(vis + base);
      vp[0] = make_float4(vz[0], vz[1], vz[2], vz[3]);
      vp[1] = make_float4(vz[4], vz[5], vz[6], vz[7]);
    } else {
#pragma unroll
      for (int k = 0; k < ITEMS; ++k) vis[base + k] = vz[k];
    }
  } else {
#pragma unroll
    for (int k = 0; k < ITEMS; ++k) {
      int i = base + k;
      if (i < n) { trans[i] = tr[k]; vis[i] = vz[k]; }
    }
  }
}

extern "C" void kernel_launch(void* const* d_in, const int* in_sizes, int n_in,
                              void* d_out, int out_size, void* d_ws, size_t ws_size,
                              hipStream_t stream) {
  (void)n_in; (void)out_size; (void)ws_size;
  const float* alphas = (const float*)d_in[0];
  const int*   ridx   = (const int*)d_in[1];
  const int n = in_sizes[0];
  const int nblocks = (n + CHUNK - 1) / CHUNK;

  float*    aggP  = (float*)d_ws;
  unsigned* aggF  = (unsigned*)(aggP + nblocks);
  float*    carry = (float*)(aggF + nblocks);

  float* trans = (float*)d_out;
  float* vis   = trans + n;

  k_aggregate<<<nblocks, BLOCK, 0, stream>>>(alphas, ridx, n, aggP, aggF);
  k_carry<<<1, BLOCK, 0, stream>>>(aggP, aggF, nblocks, carry);
  k_final<<<nblocks, BLOCK, 0, stream>>>(alphas, ridx, n, carry, trans, vis);
}